// MeanStdMemory_84275848282473
// MI455X (gfx1250) — compile-verified
//
#include <hip/hip_runtime.h>
#include <math.h>

typedef __attribute__((ext_vector_type(2))) float v2f;
typedef __attribute__((ext_vector_type(8))) float v8f;

#define B_   16
#define N_   2048
#define D_   128
#define S_   32768
#define K_   50

// ---------------- Kernel 1: per-batch mean/std + squared norms ----------------
__global__ __launch_bounds__(256) void stats_kernel(
    const float* __restrict__ fts, float* __restrict__ wmean,
    float* __restrict__ wstd, float* __restrict__ normM, float* __restrict__ normS)
{
    const int b   = blockIdx.x;
    const int tid = threadIdx.x;
    const int d    = tid & (D_ - 1);
    const int half = tid >> 7;

    const float* p = fts + (size_t)b * N_ * D_ + d;
    float s = 0.f, s2 = 0.f;
    const int n0 = half * (N_ / 2);
    for (int n = n0; n < n0 + N_ / 2; ++n) {
        float v = p[(size_t)n * D_];
        s += v;
        s2 = fmaf(v, v, s2);
    }
    __shared__ float shs[256], shq[256];
    shs[tid] = s; shq[tid] = s2;
    __syncthreads();

    float m = 0.f, sd = 0.f;
    if (tid < D_) {
        float ts = shs[tid] + shs[tid + D_];
        float tq = shq[tid] + shq[tid + D_];
        m = ts * (1.0f / N_);
        float var = tq * (1.0f / N_) - m * m;            // population std (ddof=0)
        sd = sqrtf(fmaxf(var, 0.f));
        wmean[b * D_ + tid] = m;
        wstd [b * D_ + tid] = sd;
    }
    __syncthreads();
    shs[tid] = (tid < D_) ? m * m  : 0.f;
    shq[tid] = (tid < D_) ? sd * sd : 0.f;
    __syncthreads();
    for (int off = 128; off > 0; off >>= 1) {
        if (tid < off) { shs[tid] += shs[tid + off]; shq[tid] += shq[tid + off]; }
        __syncthreads();
    }
    if (tid == 0) { normM[b] = shs[0]; normS[b] = shq[0]; }
}

// ---------------- Kernel 2: distance matrix via V_WMMA_F32_16X16X4_F32 --------
// ds[b,s] = sqrt(|means_s|^2 - 2<means_s,mean_b> + |mean_b|^2)
//         + sqrt(|stds_s|^2  - 2<stds_s, std_b>  + |std_b|^2)
// A[m,k] = stats[m][k] (m=b), B[k,n] = mem[s0+n][k].  One wave per 16-wide S tile.
__global__ __launch_bounds__(256) void dist_kernel(
    const float* __restrict__ means, const float* __restrict__ stds,
    const float* __restrict__ wmean, const float* __restrict__ wstd,
    const float* __restrict__ normM, const float* __restrict__ normS,
    float* __restrict__ ds)
{
    const int lane = threadIdx.x & 31;
    const int wave = threadIdx.x >> 5;
    const int s0   = (blockIdx.x * 8 + wave) * 16;
    const int row    = lane & 15;          // A row (=b) / B col (=s_local) source row
    const int colOff = (lane >> 4) << 1;   // K sub-offset: 0 for lanes 0-15, 2 for 16-31

    const float* am = wmean + row * D_ + colOff;
    const float* as = wstd  + row * D_ + colOff;
    const float* bm = means + (size_t)(s0 + row) * D_ + colOff;
    const float* bs = stds  + (size_t)(s0 + row) * D_ + colOff;

    v8f accM = {};
    v8f accS = {};
    float pm2 = 0.f, ps2 = 0.f;            // partial row norms of this tile's rows
#pragma unroll
    for (int k0 = 0; k0 < D_; k0 += 4) {
        v2f aM = *(const v2f*)(am + k0);
        v2f bM = *(const v2f*)(bm + k0);
        v2f aS = *(const v2f*)(as + k0);
        v2f bS = *(const v2f*)(bs + k0);
        pm2 = fmaf(bM.x, bM.x, fmaf(bM.y, bM.y, pm2));
        ps2 = fmaf(bS.x, bS.x, fmaf(bS.y, bS.y, ps2));
        // (neg_a, A, neg_b, B, c_mod, C, reuse_a, reuse_b)
        accM = __builtin_amdgcn_wmma_f32_16x16x4_f32(false, aM, false, bM,
                                                     (short)0, accM, false, false);
        accS = __builtin_amdgcn_wmma_f32_16x16x4_f32(false, aS, false, bS,
                                                     (short)0, accS, false, false);
    }
    // each lane holds half of its row's K range; combine with the partner half
    pm2 += __shfl_xor(pm2, 16, 32);
    ps2 += __shfl_xor(ps2, 16, 32);
    // now pm2/ps2 == |means[s0+row]|^2 / |stds[s0+row]|^2, and row == D-matrix N index

    const int hi = lane >> 4;
#pragma unroll
    for (int r = 0; r < 8; ++r) {
        int bb = r + (hi << 3);            // D-matrix M index (= batch)
        float dm2  = normM[bb] - 2.f * accM[r] + pm2;
        float dsd2 = normS[bb] - 2.f * accS[r] + ps2;
        float dist = sqrtf(fmaxf(dm2, 0.f)) + sqrtf(fmaxf(dsd2, 0.f));
        ds[(size_t)bb * S_ + s0 + row] = dist;
    }
}

// ---------------- Kernel 3: top-50 + softmax weights + affine coefficients ----
__global__ __launch_bounds__(1024) void topk_kernel(
    const float* __restrict__ ds, const float* __restrict__ means,
    const float* __restrict__ stds, const float* __restrict__ wmean,
    const float* __restrict__ wstd, const float* __restrict__ temp,
    const float* __restrict__ flerp, float* __restrict__ scale,
    float* __restrict__ bias)
{
    const int b   = blockIdx.x;
    const int tid = threadIdx.x;

    float v[32];
    const float* row = ds + (size_t)b * S_;
#pragma unroll
    for (int j = 0; j < 32; ++j) v[j] = row[tid + j * 1024];
    unsigned usedMask = 0u;

    __shared__ float rv[1024];
    __shared__ int   ri[1024];
    __shared__ float topv[K_];
    __shared__ int   topi[K_];
    __shared__ float wts[K_];

    for (int k = 0; k < K_; ++k) {
        float bv = 3.402823e38f; int bi = 0x7FFFFFFF;
#pragma unroll
        for (int j = 0; j < 32; ++j) {
            if (!((usedMask >> j) & 1u)) {
                float val = v[j]; int idx = tid + j * 1024;
                if (val < bv || (val == bv && idx < bi)) { bv = val; bi = idx; }
            }
        }
        rv[tid] = bv; ri[tid] = bi;
        __syncthreads();
        for (int off = 512; off > 0; off >>= 1) {
            if (tid < off) {
                float ov = rv[tid + off]; int oi = ri[tid + off];
                if (ov < rv[tid] || (ov == rv[tid] && oi < ri[tid])) {
                    rv[tid] = ov; ri[tid] = oi;
                }
            }
            __syncthreads();
        }
        int win = ri[0];
        if (tid == 0) { topv[k] = rv[0]; topi[k] = win; }
        if ((win & 1023) == tid) usedMask |= 1u << (win >> 10);  // mask out winner
        __syncthreads();
    }

    if (tid == 0) {
        float t = temp[0];
        float e[K_];
        float mx = -3.402823e38f;
        for (int k = 0; k < K_; ++k) { e[k] = expf(-t * topv[k]); mx = fmaxf(mx, e[k]); }
        float sum = 0.f;
        for (int k = 0; k < K_; ++k) { float w = expf(e[k] - mx); wts[k] = w; sum += w; }
        float inv = 1.f / sum;
        for (int k = 0; k < K_; ++k) wts[k] *= inv;
    }
    __syncthreads();

    if (tid < D_) {
        float mg = 0.f, sg = 0.f;
        for (int k = 0; k < K_; ++k) {
            int s = topi[k]; float w = wts[k];
            mg = fmaf(w, means[(size_t)s * D_ + tid], mg);
            sg = fmaf(w, stds [(size_t)s * D_ + tid], sg);
        }
        float mb = wmean[b * D_ + tid], sb = wstd[b * D_ + tid];
        float lp = 1.f / (1.f + expf(-flerp[0]));
        float mf = lp * mg + (1.f - lp) * mb;
        float sf = lp * sg + (1.f - lp) * sb;
        float sc = sf / sb;
        scale[b * D_ + tid] = sc;
        bias [b * D_ + tid] = mf - mb * sc;   // out = fts*scale + bias
    }
}

// ---------------- Kernel 4: elementwise apply (float4, streaming) -------------
__global__ __launch_bounds__(256) void apply_kernel(
    const float4* __restrict__ fts, const float4* __restrict__ scale,
    const float4* __restrict__ bias, float4* __restrict__ out)
{
    size_t i = (size_t)blockIdx.x * blockDim.x + threadIdx.x;  // 1,048,576 float4
    int d4 = (int)(i & 31);          // 32 float4 per 128-dim row
    int b  = (int)(i >> 16);         // 2048*32 float4 per batch
    float4 v  = fts[i];
    float4 sc = scale[b * 32 + d4];
    float4 bi = bias [b * 32 + d4];
    float4 o;
    o.x = fmaf(v.x, sc.x, bi.x);
    o.y = fmaf(v.y, sc.y, bi.y);
    o.z = fmaf(v.z, sc.z, bi.z);
    o.w = fmaf(v.w, sc.w, bi.w);
    out[i] = o;
}

extern "C" void kernel_launch(void* const* d_in, const int* in_sizes, int n_in,
                              void* d_out, int out_size, void* d_ws, size_t ws_size,
                              hipStream_t stream) {
    const float* node_fts   = (const float*)d_in[0];
    const float* means      = (const float*)d_in[1];
    const float* stds       = (const float*)d_in[2];
    const float* temp       = (const float*)d_in[3];
    const float* fixed_lerp = (const float*)d_in[4];
    float* out = (float*)d_out;
    float* ws  = (float*)d_ws;

    // workspace layout (floats)
    float* wmean = ws;              // 2048
    float* wstd  = ws + 2048;       // 2048
    float* normM = ws + 4096;       // 16
    float* normS = ws + 4112;       // 16
    float* scale = ws + 4128;       // 2048 (16B aligned)
    float* bias  = ws + 6176;       // 2048 (16B aligned)
    float* ds    = ws + 8224;       // 16*32768 = 524288  (~2.1 MB total)

    stats_kernel<<<B_, 256, 0, stream>>>(node_fts, wmean, wstd, normM, normS);
    dist_kernel<<<S_ / (16 * 8), 256, 0, stream>>>(means, stds, wmean, wstd,
                                                   normM, normS, ds);
    topk_kernel<<<B_, 1024, 0, stream>>>(ds, means, stds, wmean, wstd,
                                         temp, fixed_lerp, scale, bias);
    apply_kernel<<<(B_ * N_ * D_ / 4) / 256, 256, 0, stream>>>(
        (const float4*)node_fts, (const float4*)scale, (const float4*)bias,
        (float4*)out);
}